// DressedQuantumNet_85761906966957
// MI455X (gfx1250) — compile-verified
//
#include <hip/hip_runtime.h>
#include <math.h>

typedef __attribute__((ext_vector_type(2))) float v2f;
typedef __attribute__((ext_vector_type(8))) float v8f;

#define NQ   4
#define QD   6
#define DIN  512
#define COUT 10
#define BLOCK 256            // 8 waves; 256 samples per block

// ---- 4-qubit real statevector helpers (all compile-time unrolled) ----
// flattened index s = b0*8 + b1*4 + b2*2 + b3 ; wire w <-> bit (3-w)
__device__ __forceinline__ void apply_ry(float* st, const int wire, const float c, const float s) {
  const int m = 1 << (3 - wire);
#pragma unroll
  for (int i = 0; i < 16; ++i) {
    if (!(i & m)) {
      const float a0 = st[i], a1 = st[i | m];
      st[i]     = c * a0 - s * a1;
      st[i | m] = s * a0 + c * a1;
    }
  }
}

__device__ __forceinline__ void apply_cnot(float* st, const int ctrl, const int tgt) {
  const int cm = 1 << (3 - ctrl), tm = 1 << (3 - tgt);
#pragma unroll
  for (int i = 0; i < 16; ++i) {
    if ((i & cm) && !(i & tm)) {
      const float t = st[i];
      st[i]      = st[i | tm];
      st[i | tm] = t;
    }
  }
}

__global__ __launch_bounds__(BLOCK) void dressed_qnet_fused(
    const float* __restrict__ in,       // [B, 512]
    const float* __restrict__ pre_w,    // [4, 512]
    const float* __restrict__ pre_b,    // [4]
    const float* __restrict__ q_params, // [24]
    const float* __restrict__ post_w,   // [10, 4]
    const float* __restrict__ post_b,   // [10]
    float* __restrict__ out)            // [B, 10]
{
  // B panel in WMMA-B fragment pair order: bb[j*16+n] = { Bt[2j][n], Bt[2j+1][n] },
  // Bt[k][n] = pre_w[n][k] for n<4 else 0.  256*16 float2 = 32 KB.
  __shared__ v2f   bb[256 * 16];
  __shared__ float ang[BLOCK * NQ];     // angles per local sample, 4 KB

  const int tid = threadIdx.x;

  // ---- stage padded B panel (uniform work across all threads) ----
#pragma unroll
  for (int i = 0; i < 16; ++i) {
    const int e = tid + i * BLOCK;      // 0..4095
    const int j = e >> 4, n = e & 15;
    v2f v; v.x = 0.0f; v.y = 0.0f;
    if (n < NQ) { v.x = pre_w[n * DIN + 2 * j]; v.y = pre_w[n * DIN + 2 * j + 1]; }
    bb[e] = v;
  }
  __syncthreads();

  // ---- phase 1: WMMA pre-GEMM, 2 tiles (32 rows) per wave ----
  const int lane = tid & 31;
  const int wave = tid >> 5;
  const int half = lane >> 4;           // 0: K lanes {0,1}, 1: K lanes {2,3}
  const int lrow = lane & 15;           // M for A, N for B/C/D
  const long rowBase = (long)blockIdx.x * BLOCK + (long)wave * 32;

  const float* pA0 = in + (rowBase + lrow) * DIN + half * 2;  // tile0 rows
  const float* pA1 = pA0 + 16 * DIN;                          // tile1 rows

  v8f c0 = {};
  v8f c1 = {};
#pragma unroll 4
  for (int k = 0; k < DIN; k += 4) {
    const v2f a0 = *(const v2f*)(pA0 + k);
    const v2f a1 = *(const v2f*)(pA1 + k);
    const v2f b  = bb[((k >> 1) + half) * 16 + lrow];
    c0 = __builtin_amdgcn_wmma_f32_16x16x4_f32(false, a0, false, b, (short)0, c0, false, false);
    c1 = __builtin_amdgcn_wmma_f32_16x16x4_f32(false, a1, false, b, (short)0, c1, false, false);
  }

  // C/D layout: lane l, vgpr v -> (M = v + 8*(l>=16), N = l&15). Keep N<4.
  if (lrow < NQ) {
    const float bias    = pre_b[lrow];
    const float HALF_PI = 1.57079632679489661923f;
#pragma unroll
    for (int v = 0; v < 8; ++v) {
      const int M = v + half * 8;
      ang[(wave * 32 + M)      * NQ + lrow] = tanhf(c0[v] + bias) * HALF_PI;
      ang[(wave * 32 + 16 + M) * NQ + lrow] = tanhf(c1[v] + bias) * HALF_PI;
    }
  }
  __syncthreads();

  // ---- phase 2: one sample per thread, 16-amp statevector in registers ----
  float st[16];
#pragma unroll
  for (int i = 0; i < 16; ++i) st[i] = 0.25f;   // H^{x4} |0000>

#pragma unroll
  for (int w = 0; w < NQ; ++w) {
    const float th = ang[tid * NQ + w] * 0.5f;
    float s, c;
    __sincosf(th, &s, &c);
    apply_ry(st, w, c, s);
  }

#pragma unroll
  for (int k = 0; k < QD; ++k) {
    apply_cnot(st, 0, 1);
    apply_cnot(st, 2, 3);
    apply_cnot(st, 1, 2);
#pragma unroll
    for (int w = 0; w < NQ; ++w) {
      const float th = q_params[k * NQ + w] * 0.5f;
      float s, c;
      __sincosf(th, &s, &c);
      apply_ry(st, w, c, s);
    }
  }

  // PauliZ expectations
  float pr[16];
#pragma unroll
  for (int i = 0; i < 16; ++i) pr[i] = st[i] * st[i];

  float qo[NQ];
#pragma unroll
  for (int w = 0; w < NQ; ++w) {
    const int m = 1 << (3 - w);
    float e = 0.0f;
#pragma unroll
    for (int i = 0; i < 16; ++i) e += (i & m) ? -pr[i] : pr[i];
    qo[w] = e;
  }

  // ---- post-net: [4] -> [10] ----
  const long sample = (long)blockIdx.x * BLOCK + tid;
  float* po = out + sample * COUT;
#pragma unroll
  for (int c = 0; c < COUT; ++c) {
    float acc = post_b[c];
#pragma unroll
    for (int w = 0; w < NQ; ++w) acc += qo[w] * post_w[c * NQ + w];
    po[c] = acc;
  }
}

extern "C" void kernel_launch(void* const* d_in, const int* in_sizes, int n_in,
                              void* d_out, int out_size, void* d_ws, size_t ws_size,
                              hipStream_t stream) {
  const float* in       = (const float*)d_in[0];
  const float* pre_w    = (const float*)d_in[1];
  const float* pre_b    = (const float*)d_in[2];
  const float* q_params = (const float*)d_in[3];
  const float* post_w   = (const float*)d_in[4];
  const float* post_b   = (const float*)d_in[5];
  float* out = (float*)d_out;

  const int B    = in_sizes[0] / DIN;   // 65536
  const int grid = B / BLOCK;           // 256 blocks of 256 threads

  dressed_qnet_fused<<<grid, BLOCK, 0, stream>>>(in, pre_w, pre_b, q_params,
                                                 post_w, post_b, out);
}